// GCNConvLayerNorm_84954453115089
// MI455X (gfx1250) — compile-verified
//
#include <hip/hip_runtime.h>
#include <hip/hip_bf16.h>

#define DIM 64
#define LN_EPSF 1e-5f
#define DEG_EPSF 1e-12f

typedef float v2f __attribute__((ext_vector_type(2)));
typedef float v8f __attribute__((ext_vector_type(8)));

// ---------------------------------------------------------------------------
// K0: zero a float buffer (grid-stride)
// ---------------------------------------------------------------------------
__global__ void gcn_zero_f32(float* __restrict__ p, size_t n) {
    size_t i = (size_t)blockIdx.x * blockDim.x + threadIdx.x;
    size_t stride = (size_t)gridDim.x * blockDim.x;
    for (; i < n; i += stride) p[i] = 0.0f;
}

// ---------------------------------------------------------------------------
// K1: deg[src[e]] += 1   (edge multiplicity counts)
// ---------------------------------------------------------------------------
__global__ void gcn_degree(const int* __restrict__ src, float* __restrict__ deg, int E) {
    int e = blockIdx.x * blockDim.x + threadIdx.x;
    if (e < E) atomicAdd(&deg[src[e]], 1.0f);
}

// ---------------------------------------------------------------------------
// K2: dinv[i] = max(deg, eps)^-1/2
// ---------------------------------------------------------------------------
__global__ void gcn_dinv(const float* __restrict__ deg, float* __restrict__ dinv, int N) {
    int i = blockIdx.x * blockDim.x + threadIdx.x;
    if (i < N) dinv[i] = rsqrtf(fmaxf(deg[i], DEG_EPSF));
}

// ---------------------------------------------------------------------------
// K3: SpMM scatter: x2[src[e], :] += x[dst[e], :] * dinv[dst[e]]
// 16 threads per edge, float4 (B128) gathers, 4 dword f32 atomics each.
// Working set lives in L2 (192 MB) -> L2-atomic bound, the dominant phase.
// ---------------------------------------------------------------------------
__global__ void gcn_scatter(const float* __restrict__ x, const int* __restrict__ src,
                            const int* __restrict__ dst, const float* __restrict__ dinv,
                            float* __restrict__ x2, int E) {
    int gid = blockIdx.x * blockDim.x + threadIdx.x;   // E*16 threads
    int e = gid >> 4;
    if (e >= E) return;
    int d4 = (gid & 15) * 4;
    int s = src[e];
    int t = dst[e];
    float dr = dinv[t];
    const float4 v = *(const float4*)(x + (size_t)t * DIM + d4);
    float* o = x2 + (size_t)s * DIM + d4;
    atomicAdd(o + 0, v.x * dr);
    atomicAdd(o + 1, v.y * dr);
    atomicAdd(o + 2, v.z * dr);
    atomicAdd(o + 3, v.w * dr);
}

// ---------------------------------------------------------------------------
// K4: fused  h = (x2 * dinv) @ W + x ;  LayerNorm(h) ; ReLU  -> io (in place)
//
// One wave (32 lanes) per 16-row tile. Full-precision V_WMMA_F32_16X16X4_F32:
//   A (16x4 fp32): lane l%16 = row M; lane half selects K {0,1} vs {2,3}
//                  -> per k-step each lane loads a contiguous float2 of its row
//   B (4x16 fp32): lane l%16 = col N; v0/v1 = K rows {0,1} (half 0) / {2,3} (half 1)
//   C/D (16x16):   8 VGPRs, lane half selects M 0..7 vs 8..15
// Residual x is preloaded into the C accumulators. LayerNorm is done fully in
// registers: per-lane partial Σh, Σh² over 4 N-tiles + shfl_xor(1,2,4,8)
// reduction inside each 16-lane half of the wave32.
// ---------------------------------------------------------------------------
__global__ __launch_bounds__(32)
void gcn_gemm_ln_relu(const float* __restrict__ xres,   // residual input [N,64]
                      const float* __restrict__ W,      // [64,64] row-major (K x N)
                      const float* __restrict__ dinv,   // [N]
                      const float* __restrict__ gamma,  // [64]
                      const float* __restrict__ beta,   // [64]
                      float* __restrict__ io,           // in: x2 accumulator, out: result [N,64]
                      int N) {
    const int lane = threadIdx.x;        // 0..31
    const int half = lane >> 4;          // 0/1
    const int l16  = lane & 15;
    const int row0 = blockIdx.x * 16;

    // --- A-side row for this lane (clamped; EXEC must stay all-ones for WMMA)
    const int arow = min(row0 + l16, N - 1);
    const float dr = dinv[arow];

    // --- C/D rows for this lane
    const int crow0 = row0 + half * 8;

    // Init accumulators with residual x (fuses "+ x")
    v8f acc[4];
#pragma unroll
    for (int nt = 0; nt < 4; ++nt) {
#pragma unroll
        for (int r = 0; r < 8; ++r) {
            int rr = min(crow0 + r, N - 1);
            acc[nt][r] = xres[(size_t)rr * DIM + nt * 16 + l16];
        }
    }

    // K loop: 16 chained K=4 fp32 WMMAs per N-tile (64 total)
#pragma unroll
    for (int kk = 0; kk < 16; ++kk) {
        const int ka = 4 * kk + 2 * half;
        v2f a;
        a.x = io[(size_t)arow * DIM + ka + 0] * dr;
        a.y = io[(size_t)arow * DIM + ka + 1] * dr;
#pragma unroll
        for (int nt = 0; nt < 4; ++nt) {
            v2f b;
            b.x = W[(ka + 0) * DIM + nt * 16 + l16];
            b.y = W[(ka + 1) * DIM + nt * 16 + l16];
            acc[nt] = __builtin_amdgcn_wmma_f32_16x16x4_f32(
                /*neg_a=*/false, a, /*neg_b=*/false, b,
                /*c_mod=*/(short)0, acc[nt], /*reuse_a=*/false, /*reuse_b=*/false);
        }
    }

    // --- LayerNorm: per-row Σh, Σh² (each row spread over 16 lanes x 4 tiles)
    float s1[8], s2[8];
#pragma unroll
    for (int r = 0; r < 8; ++r) {
        float a1 = 0.0f, a2 = 0.0f;
#pragma unroll
        for (int nt = 0; nt < 4; ++nt) {
            float v = acc[nt][r];
            a1 += v;
            a2 += v * v;
        }
        s1[r] = a1;
        s2[r] = a2;
    }
#pragma unroll
    for (int m = 1; m <= 8; m <<= 1) {
#pragma unroll
        for (int r = 0; r < 8; ++r) {
            s1[r] += __shfl_xor(s1[r], m, 32);
            s2[r] += __shfl_xor(s2[r], m, 32);
        }
    }

    // --- normalize + gamma/beta + ReLU, store (in place; all reads of these
    //     rows completed above)
#pragma unroll
    for (int nt = 0; nt < 4; ++nt) {
        const int col = nt * 16 + l16;
        const float g  = gamma[col];
        const float bb = beta[col];
#pragma unroll
        for (int r = 0; r < 8; ++r) {
            const int rr = crow0 + r;
            if (rr < N) {
                float mean = s1[r] * (1.0f / 64.0f);
                float var  = s2[r] * (1.0f / 64.0f) - mean * mean;
                float rstd = rsqrtf(var + LN_EPSF);
                float v = (acc[nt][r] - mean) * rstd * g + bb;
                io[(size_t)rr * DIM + col] = fmaxf(v, 0.0f);
            }
        }
    }
}

// ---------------------------------------------------------------------------
// Launch
// ---------------------------------------------------------------------------
extern "C" void kernel_launch(void* const* d_in, const int* in_sizes, int n_in,
                              void* d_out, int out_size, void* d_ws, size_t ws_size,
                              hipStream_t stream) {
    const float* x     = (const float*)d_in[0];
    const int*   src   = (const int*)d_in[1];
    const int*   dst   = (const int*)d_in[2];
    const float* W     = (const float*)d_in[3];
    const float* gamma = (const float*)d_in[4];
    const float* beta  = (const float*)d_in[5];
    float* out = (float*)d_out;

    const int N = in_sizes[0] / DIM;   // 100000
    const int E = in_sizes[1];         // 1200000

    float* deg  = (float*)d_ws;        // [N]
    float* dinv = deg + N;             // [N]

    // zero accumulator (d_out doubles as x2) and degree buffer
    gcn_zero_f32<<<4096, 256, 0, stream>>>(out, (size_t)N * DIM);
    gcn_zero_f32<<<(N + 255) / 256, 256, 0, stream>>>(deg, (size_t)N);

    gcn_degree<<<(E + 255) / 256, 256, 0, stream>>>(src, deg, E);
    gcn_dinv<<<(N + 255) / 256, 256, 0, stream>>>(deg, dinv, N);

    // scatter: 16 threads per edge
    {
        long long threads = (long long)E * 16;
        int blocks = (int)((threads + 255) / 256);
        gcn_scatter<<<blocks, 256, 0, stream>>>(x, src, dst, dinv, out, E);
    }

    // fused GEMM + residual + LayerNorm + ReLU: one wave per 16 rows
    gcn_gemm_ln_relu<<<(N + 15) / 16, 32, 0, stream>>>(x, W, dinv, gamma, beta, out, N);
}